// Agent_19258633355360
// MI455X (gfx1250) — compile-verified
//
#include <hip/hip_runtime.h>

typedef float v2f __attribute__((ext_vector_type(2)));
typedef float v8f __attribute__((ext_vector_type(8)));

#define TSTEPS 1024
#define BATCH  512
#define NIN    50
#define NH     10
#define NG     40          // 4*H gates
#define MT     8           // M-tiles per wave in the GEMM kernels

// ---------------------------------------------------------------------------
// Kernel 1: xg[row][g] = sum_i x[row][i]*W_ih[g][i] + b_ih[g] + b_hh[g]
// rows = T*B = 524288.  fp32 WMMA 16x16x4, K=50 -> 13 chunks, N=40 -> 3 tiles.
// Zero-padded B panel (W_ih^T, WMMA operand layout) staged in LDS per block.
// __syncthreads() inside the M-tile loop stops LICM from hoisting the 39
// ds-loads back into registers (which caused scratch spills).  Inner loop:
// 1 global b64 A-load + 3 ds b64 B-loads + 3 WMMAs per K-chunk, ~40 live VGPRs.
// ---------------------------------------------------------------------------
__global__ __launch_bounds__(256)
void k_xg(const float* __restrict__ x,
          const float* __restrict__ W_ih,
          const float* __restrict__ b_ih,
          const float* __restrict__ b_hh,
          float* __restrict__ xg) {
  __shared__ v2f Blds[13 * 3 * 32];          // 9984 bytes

  const int lane = threadIdx.x & 31;
  const int wave = threadIdx.x >> 5;
  const int lm   = lane & 15;
  const int hi   = lane >> 4;                // K-pair select: {0,1} vs {2,3}
  const int wid  = blockIdx.x * 8 + wave;
  const int mbase0 = wid * (16 * MT);

  // ---- fill LDS with zero-padded B = W_ih^T in per-lane operand layout ----
  for (int e = threadIdx.x; e < 13 * 3 * 32; e += 256) {
    const int l  = e & 31;
    const int kt = e >> 5;                   // 0..38
    const int kk = kt / 3, t = kt % 3;
    const int lm_ = l & 15, hi_ = l >> 4;
    const int k0 = kk * 4 + hi_ * 2, k1 = k0 + 1;
    const int n  = t * 16 + lm_;
    const unsigned kc0 = (k0 < NIN) ? k0 : 0u;
    const unsigned kc1 = (k1 < NIN) ? k1 : 0u;
    const unsigned nc  = (n  < NG ) ? n  : 0u;
    const float v0 = W_ih[nc * NIN + kc0];   // unconditional clamped load
    const float v1 = W_ih[nc * NIN + kc1];
    v2f v;
    v.x = (k0 < NIN && n < NG) ? v0 : 0.0f;
    v.y = (k1 < NIN && n < NG) ? v1 : 0.0f;
    Blds[e] = v;
  }

  // ---- per-column bias (splat into accumulator init) ----------------------
  float bias[3];
#pragma unroll
  for (int t = 0; t < 3; ++t) {
    const int n  = t * 16 + lm;
    const unsigned nc = (n < NG) ? n : 0u;
    const float bv = b_ih[nc] + b_hh[nc];
    bias[t] = (n < NG) ? bv : 0.0f;
  }

  // ---- M-tile loop --------------------------------------------------------
#pragma unroll 1
  for (int mt = 0; mt < MT; ++mt) {
    __syncthreads();                         // fence: keep ds-loads in-loop
    const int mbase = mbase0 + mt * 16;
    const unsigned abase = (unsigned)(mbase + lm) * NIN;   // 32-bit offset

    v8f acc[3];
#pragma unroll
    for (int t = 0; t < 3; ++t)
#pragma unroll
      for (int r = 0; r < 8; ++r) acc[t][r] = bias[t];

#pragma unroll
    for (int kk = 0; kk < 13; ++kk) {
      v2f a;
      if (kk < 12) {                         // compile-time branch (unrolled)
        a = *(const v2f*)(x + abase + kk * 4 + hi * 2);
      } else {                               // K-tail: k = 48..51
        const v2f tail = *(const v2f*)(x + abase + 48);
        a.x = hi ? 0.0f : tail.x;
        a.y = hi ? 0.0f : tail.y;
      }
      const v2f b0 = Blds[(kk * 3 + 0) * 32 + lane];
      const v2f b1 = Blds[(kk * 3 + 1) * 32 + lane];
      const v2f b2 = Blds[(kk * 3 + 2) * 32 + lane];
      acc[0] = __builtin_amdgcn_wmma_f32_16x16x4_f32(
          false, a, false, b0, (short)0, acc[0], false, false);
      acc[1] = __builtin_amdgcn_wmma_f32_16x16x4_f32(
          false, a, false, b1, (short)0, acc[1], false, false);
      acc[2] = __builtin_amdgcn_wmma_f32_16x16x4_f32(
          false, a, false, b2, (short)0, acc[2], false, false);
    }

#pragma unroll
    for (int t = 0; t < 3; ++t) {
      const int n = t * 16 + lm;
      if (n < NG) {
        const unsigned sbase = (unsigned)(mbase + 8 * hi) * NG + n;
#pragma unroll
        for (int r = 0; r < 8; ++r)          // r*NG folds into imm offset
          xg[sbase + r * NG] = acc[t][r];
      }
    }
  }
}

// ---------------------------------------------------------------------------
// Kernel 2: latency-optimized LSTM recurrence (the serial critical path).
// 4 lanes per batch element: lane r owns gate type r (i/f/g/o), 10 units each
// (100 FMAs/lane/step). Quad exchange via ds_bpermute; each lane keeps the
// full c[10]/h[10]. xg loads for step t+1 issued at the top of step t
// (xg is L2-resident: 84 MB < 192 MB L2).  Only 64 waves exist on the whole
// device, so occupancy is irrelevant: waves_per_eu(1) grants the allocator a
// full VGPR budget (W_hh slice alone is 100 VGPRs) -> no spills on the chain.
// ---------------------------------------------------------------------------
__global__ __launch_bounds__(256)
__attribute__((amdgpu_waves_per_eu(1)))
void k_lstm(const float* __restrict__ xg,
            const float* __restrict__ W_hh,
            float* __restrict__ hs) {
  const int tid = blockIdx.x * blockDim.x + threadIdx.x;
  const int b = tid >> 2;                    // batch element
  const int r = tid & 3;                     // gate type: 0=i 1=f 2=g 3=o
  const int lane = threadIdx.x & 31;
  const int qb = (lane & ~3) * 4;            // byte addr of quad lane 0

  float W[10][10];
#pragma unroll
  for (int j = 0; j < 10; ++j)
#pragma unroll
    for (int k = 0; k < 10; ++k)
      W[j][k] = W_hh[(unsigned)(r * 10 + j) * 10 + k];

  // Unified activation: act = Aa/(1+exp(-sc*x)) + Bb  (sigmoid / tanh)
  const float sc = (r == 2) ? 2.0f : 1.0f;
  const float Aa = (r == 2) ? 2.0f : 1.0f;
  const float Bb = (r == 2) ? -1.0f : 0.0f;

  float h[10], c[10];
#pragma unroll
  for (int j = 0; j < 10; ++j) { h[j] = 0.0f; c[j] = 0.0f; }

  const unsigned xbase = (unsigned)b * NG + (unsigned)r * 10;

  v2f xv[5];
#pragma unroll
  for (int q = 0; q < 5; ++q) xv[q] = *(const v2f*)(xg + xbase + 2 * q);

  for (int t = 0; t < TSTEPS; ++t) {
    // Issue next step's loads immediately (clamped, branchless).
    const unsigned tn = (t + 1 < TSTEPS) ? (unsigned)(t + 1) : (unsigned)t;
    const unsigned xoff = xbase + tn * (BATCH * NG);
    v2f xn[5];
#pragma unroll
    for (int q = 0; q < 5; ++q) xn[q] = *(const v2f*)(xg + xoff + 2 * q);

    float act[10];
#pragma unroll
    for (int j = 0; j < 10; ++j) {
      float pre = (j & 1) ? xv[j >> 1].y : xv[j >> 1].x;
#pragma unroll
      for (int k = 0; k < 10; ++k) pre += W[j][k] * h[k];
      const float e = __expf(-sc * pre);
      act[j] = Aa / (1.0f + e) + Bb;
    }

#pragma unroll
    for (int j = 0; j < 10; ++j) {
      const int av = __float_as_int(act[j]);
      const float iv = __int_as_float(__builtin_amdgcn_ds_bpermute(qb + 0,  av));
      const float fv = __int_as_float(__builtin_amdgcn_ds_bpermute(qb + 4,  av));
      const float gv = __int_as_float(__builtin_amdgcn_ds_bpermute(qb + 8,  av));
      const float ov = __int_as_float(__builtin_amdgcn_ds_bpermute(qb + 12, av));
      c[j] = fv * c[j] + iv * gv;
      const float e2 = __expf(-2.0f * c[j]);
      const float th = 2.0f / (1.0f + e2) - 1.0f;
      h[j] = ov * th;
    }

    if (r == 0) {
      const unsigned hoff = ((unsigned)t * BATCH + (unsigned)b) * NH;
#pragma unroll
      for (int q = 0; q < 5; ++q) {
        v2f v; v.x = h[2 * q]; v.y = h[2 * q + 1];
        *(v2f*)(hs + hoff + 2 * q) = v;
      }
    }

#pragma unroll
    for (int q = 0; q < 5; ++q) xv[q] = xn[q];
  }
}

// ---------------------------------------------------------------------------
// Kernel 3: out[row][n] = sum_j hs[row][j] * W_out[n][j]
// rows = T*B, K=10 -> 3 chunks, N=50 -> 4 tiles.  Same LDS-B structure.
// ---------------------------------------------------------------------------
__global__ __launch_bounds__(256)
void k_out(const float* __restrict__ hs,
           const float* __restrict__ W_out,
           float* __restrict__ out) {
  __shared__ v2f Blds[3 * 4 * 32];           // 3072 bytes

  const int lane = threadIdx.x & 31;
  const int wave = threadIdx.x >> 5;
  const int lm   = lane & 15;
  const int hi   = lane >> 4;
  const int wid  = blockIdx.x * 8 + wave;
  const int mbase0 = wid * (16 * MT);

  for (int e = threadIdx.x; e < 3 * 4 * 32; e += 256) {
    const int l  = e & 31;
    const int kt = e >> 5;                   // 0..11
    const int kk = kt / 4, t = kt % 4;
    const int lm_ = l & 15, hi_ = l >> 4;
    const int k0 = kk * 4 + hi_ * 2, k1 = k0 + 1;
    const int n  = t * 16 + lm_;
    const unsigned kc0 = (k0 < NH)  ? k0 : 0u;
    const unsigned kc1 = (k1 < NH)  ? k1 : 0u;
    const unsigned nc  = (n  < NIN) ? n  : 0u;
    const float v0 = W_out[nc * NH + kc0];
    const float v1 = W_out[nc * NH + kc1];
    v2f v;
    v.x = (k0 < NH && n < NIN) ? v0 : 0.0f;
    v.y = (k1 < NH && n < NIN) ? v1 : 0.0f;
    Blds[e] = v;
  }

#pragma unroll 1
  for (int mt = 0; mt < MT; ++mt) {
    __syncthreads();                         // fence: keep ds-loads in-loop
    const int mbase = mbase0 + mt * 16;
    const unsigned abase = (unsigned)(mbase + lm) * NH;

    v2f A[3];
    A[0] = *(const v2f*)(hs + abase + hi * 2);       // k = {0,1} / {2,3}
    A[1] = *(const v2f*)(hs + abase + 4 + hi * 2);   // k = {4,5} / {6,7}
    {                                                // K-tail: k = 8..11
      const v2f tail = *(const v2f*)(hs + abase + 8);
      A[2].x = hi ? 0.0f : tail.x;
      A[2].y = hi ? 0.0f : tail.y;
    }

    v8f acc[4] = {};
#pragma unroll
    for (int kk = 0; kk < 3; ++kk) {
#pragma unroll
      for (int t = 0; t < 4; ++t) {
        const v2f bb = Blds[(kk * 4 + t) * 32 + lane];
        acc[t] = __builtin_amdgcn_wmma_f32_16x16x4_f32(
            false, A[kk], false, bb, (short)0, acc[t], false, false);
      }
    }

#pragma unroll
    for (int t = 0; t < 4; ++t) {
      const int n = t * 16 + lm;
      if (n < NIN) {
        const unsigned sbase = (unsigned)(mbase + 8 * hi) * NIN + n;
#pragma unroll
        for (int r = 0; r < 8; ++r)          // r*NIN folds into imm offset
          out[sbase + r * NIN] = acc[t][r];
      }
    }
  }
}

// ---------------------------------------------------------------------------
extern "C" void kernel_launch(void* const* d_in, const int* in_sizes, int n_in,
                              void* d_out, int out_size, void* d_ws, size_t ws_size,
                              hipStream_t stream) {
  const float* x     = (const float*)d_in[0];
  const float* W_ih  = (const float*)d_in[1];
  const float* W_hh  = (const float*)d_in[2];
  const float* b_ih  = (const float*)d_in[3];
  const float* b_hh  = (const float*)d_in[4];
  const float* W_out = (const float*)d_in[5];
  float* out = (float*)d_out;

  // Stage xg (T*B*40 = 20.97M floats) inside d_out (26.2M floats):
  // K1 writes it, K2 consumes it, K3 then overwrites d_out with the result.
  float* xg = out;
  float* hs = (float*)d_ws;                  // T*B*10 floats = 21 MB

  const int rows   = TSTEPS * BATCH;          // 524288
  const int mtiles = rows / 16;               // 32768
  const int waves  = mtiles / MT;             // 4096
  const int blocks = waves / 8;               // 512 (8 waves/block)

  k_xg  <<<blocks, 256, 0, stream>>>(x, W_ih, b_ih, b_hh, xg);
  k_lstm<<<(BATCH * 4) / 256, 256, 0, stream>>>(xg, W_hh, hs);
  k_out <<<blocks, 256, 0, stream>>>(hs, W_out, out);
}